// MeanAggregator_74131135529475
// MI455X (gfx1250) — compile-verified
//
#include <hip/hip_runtime.h>

// CDNA5 / gfx1250 — mean neighbor aggregation via V_WMMA_F32_16X16X4_F32.
// One wave32 per node: out[i,:] = (1/cnt_i) * sum_k mask[i,k] * feat[idx[i,k],:]
//
//  A[16x4]  = mask chunk (4 neighbors) broadcast to all 16 rows
//  B[4x16]  = 4 gathered feature rows x 16 feature columns
//  C[16x16] accumulated over 16 K-chunks (64 neighbors) in full f32.
// All C rows are identical (A rows identical); lanes 0-15 / VGPR0 hold the row.

typedef float v2f __attribute__((ext_vector_type(2)));
typedef float v8f __attribute__((ext_vector_type(8)));

#define D_FEAT 128
#define N_NEIGH 64
#define WAVES_PER_BLOCK 8

__global__ __launch_bounds__(256)
void MeanAggregator_74131135529475_kernel(const float* __restrict__ feat,
                                          const int*   __restrict__ nidx,
                                          const int*   __restrict__ nmask,
                                          float*       __restrict__ out,
                                          int n_nodes)
{
    const int lane = threadIdx.x & 31;
    const int wave = threadIdx.x >> 5;
    const int node = blockIdx.x * WAVES_PER_BLOCK + wave;
    // wave-uniform exit: EXEC stays all-1s for surviving waves (WMMA requirement)
    if (node >= n_nodes) return;

    const int* ip = nidx  + (size_t)node * N_NEIGH;
    const int* mp = nmask + (size_t)node * N_NEIGH;

    // 64 neighbor indices / masks live in 2 VGPRs per lane
    const int idx_lo = ip[lane];
    const int idx_hi = ip[lane + 32];
    const int mb_lo  = (mp[lane]      != 0) ? 1 : 0;
    const int mb_hi  = (mp[lane + 32] != 0) ? 1 : 0;
    const float fm_lo = (float)mb_lo;
    const float fm_hi = (float)mb_hi;

    // valid-neighbor count via wave32 ballots
    const unsigned long long b0 = __ballot(mb_lo);
    const unsigned long long b1 = __ballot(mb_hi);
    const int   cnt = __popcll(b0) + __popcll(b1);
    const float inv = (cnt > 0) ? (1.0f / (float)cnt) : 0.0f;

    // Warm L2/L0 with every row this wave will gather (global_prefetch_b8)
    __builtin_prefetch(feat + (size_t)idx_lo * D_FEAT, 0, 1);
    __builtin_prefetch(feat + (size_t)idx_hi * D_FEAT, 0, 1);

    const int colL = lane & 15;           // feature column within 16-wide tile
    const int off0 = (lane & 16) ? 2 : 0; // K-offset within 4-chunk (half-wave split)

    // 8 accumulators: one 16x16 f32 C tile per 16-column feature tile
    v8f acc[8];
#pragma unroll
    for (int dt = 0; dt < 8; ++dt) acc[dt] = (v8f){0.f,0.f,0.f,0.f,0.f,0.f,0.f,0.f};

#pragma unroll 4
    for (int kc = 0; kc < 16; ++kc) {
        const int  base = kc << 2;          // first neighbor of this K=4 chunk
        const int  kA   = base + off0;      // neighbor feeding VGPR0 of A/B
        const int  kB   = kA + 1;           // neighbor feeding VGPR1 of A/B
        const bool hi   = (base >= 32);     // wave-uniform chunk-half select
        const int  srcA = kA & 31;
        const int  srcB = kB & 31;

        const int   iA = __shfl(hi ? idx_hi : idx_lo, srcA);
        const int   iB = __shfl(hi ? idx_hi : idx_lo, srcB);
        const float mA = __shfl(hi ? fm_hi  : fm_lo,  srcA);
        const float mB = __shfl(hi ? fm_hi  : fm_lo,  srcB);

        v2f a; a.x = mA; a.y = mB;          // A rows all identical -> C rows identical

        const float* rA = feat + (size_t)iA * D_FEAT + colL;
        const float* rB = feat + (size_t)iB * D_FEAT + colL;

#pragma unroll
        for (int dt = 0; dt < 8; ++dt) {
            v2f b; b.x = rA[dt * 16]; b.y = rB[dt * 16];   // gathered B[4x16] slice
            acc[dt] = __builtin_amdgcn_wmma_f32_16x16x4_f32(
                false, a, false, b, (short)0, acc[dt], false, false);
        }
    }

    // Row 0 of each C tile lives in VGPR0 of lanes 0-15
    float* op = out + (size_t)node * D_FEAT + colL;
    if (lane < 16) {
#pragma unroll
        for (int dt = 0; dt < 8; ++dt)
            op[dt * 16] = acc[dt].x * inv;
    }
}

extern "C" void kernel_launch(void* const* d_in, const int* in_sizes, int n_in,
                              void* d_out, int out_size, void* d_ws, size_t ws_size,
                              hipStream_t stream) {
    (void)n_in; (void)out_size; (void)d_ws; (void)ws_size;
    const float* feat  = (const float*)d_in[0];   // [60000, 128] f32
    const int*   nidx  = (const int*)  d_in[1];   // [10000, 64] int
    const int*   nmask = (const int*)  d_in[2];   // [10000, 64] int (bool-valued)
    float*       out   = (float*)      d_out;     // [10000, 128] f32

    const int n_nodes = in_sizes[1] / N_NEIGH;
    const int blocks  = (n_nodes + WAVES_PER_BLOCK - 1) / WAVES_PER_BLOCK;

    MeanAggregator_74131135529475_kernel<<<blocks, 256, 0, stream>>>(
        feat, nidx, nmask, out, n_nodes);
}